// PerspectiveTransformerLayer_6579889897980
// MI455X (gfx1250) — compile-verified
//
#include <hip/hip_runtime.h>

typedef float v2f __attribute__((ext_vector_type(2)));
typedef float v8f __attribute__((ext_vector_type(8)));

#define BB 8
#define CC 64
#define HP 256
#define WP 512
#define HB 256
#define WB 256
#define PV_PLANE (HP * WP)   // 131072
#define OUT_PLANE (HB * WB)  // 65536

__global__ __launch_bounds__(256)
void perspective_warp_kernel(const float* __restrict__ pv,
                             const float* __restrict__ rx,
                             const float* __restrict__ ry,
                             const float* __restrict__ rz,
                             float* __restrict__ out)
{
    const int tid  = threadIdx.x;
    const int lane = tid & 31;
    const unsigned gidx = blockIdx.x * 256u + tid;   // pixel id in [0, 8*256*256)
    const int b  = gidx >> 16;         // batch
    const int h  = (gidx >> 8) & 255;  // output row (uniform per block)
    const int w  = gidx & 255;         // output col (lane-contiguous)
    const int w0 = w & ~31;            // wave-uniform base column

    // ---------------- homography H = K (R + t n^T) K^-1 (per batch, cheap) --------------
    float sxv, cxv; sincosf(rx[b], &sxv, &cxv);
    float syv, cyv; sincosf(ry[b], &syv, &cyv);
    float szv, czv; sincosf(rz[b], &szv, &czv);

    // M = Ry @ Rz
    const float M00 = cyv * czv, M01 = -cyv * szv, M02 = syv;
    const float M10 = szv,       M11 = czv,        M12 = 0.0f;
    const float M20 = -syv * czv, M21 = syv * szv, M22 = cyv;
    // R = Rx @ M
    const float R00 = M00, R01 = M01, R02 = M02;
    const float R10 = cxv * M10 - sxv * M20, R11 = cxv * M11 - sxv * M21, R12 = cxv * M12 - sxv * M22;
    const float R20 = sxv * M10 + cxv * M20, R21 = sxv * M11 + cxv * M21, R22 = sxv * M12 + cxv * M22;

    // bv pivot in normalized coords: K^-1 [128,256,1] = [-0.256, 0.256, 1]; pv pivot = [0, 0.256, 1]
    const float Rp0 = R00 * -0.256f + R01 * 0.256f + R02;
    const float Rp1 = R10 * -0.256f + R11 * 0.256f + R12;
    const float Rp2 = R20 * -0.256f + R21 * 0.256f + R22;
    // s = Rp2 (tz=0, pv_piv.z=1); t = s*pv_piv - Rp; t2 = 0
    const float t0 = -Rp0;
    const float t1 = 0.256f * Rp2 - Rp1;
    // Hmat = R with third column += t
    const float G02 = R02 + t0, G12 = R12 + t1, G22 = R22;
    // KH = K @ Hmat
    const float KH00 = 500.f * R00 + 256.f * R20, KH01 = 500.f * R01 + 256.f * R21, KH02 = 500.f * G02 + 256.f * G22;
    const float KH10 = 500.f * R10 + 128.f * R20, KH11 = 500.f * R11 + 128.f * R21, KH12 = 500.f * G12 + 128.f * G22;
    const float KH20 = R20,                      KH21 = R21,                        KH22 = G22;
    // H = KH @ K^-1 ;  K^-1 = [[.002,0,-.512],[0,.002,-.256],[0,0,1]]
    const float h00 = KH00 * 0.002f, h01 = KH01 * 0.002f, h02 = -0.512f * KH00 - 0.256f * KH01 + KH02;
    const float h10 = KH10 * 0.002f, h11 = KH11 * 0.002f, h12 = -0.512f * KH10 - 0.256f * KH11 + KH12;
    const float h20 = KH20 * 0.002f, h21 = KH21 * 0.002f, h22 = -0.512f * KH20 - 0.256f * KH21 + KH22;

    // ---------------- coords = H @ [gx, gy, 1] via V_WMMA_F32_16X16X4_F32 ----------------
    // A (16x4): rows 0..2 = H rows, rest zero.  ISA layout: v0 = K0(lanes0-15)/K2(lanes16-31),
    //                                           v1 = K1/K3, M = lane % 16.
    const int r = lane & 15;
    const float Hr0 = r == 0 ? h00 : (r == 1 ? h10 : (r == 2 ? h20 : 0.0f));
    const float Hr1 = r == 0 ? h01 : (r == 1 ? h11 : (r == 2 ? h21 : 0.0f));
    const float Hr2 = r == 0 ? h02 : (r == 1 ? h12 : (r == 2 ? h22 : 0.0f));
    const bool hi = lane >= 16;
    v2f a;
    a.x = hi ? Hr2 : Hr0;   // K=2 column is the '1' multiplier slot
    a.y = hi ? 0.0f : Hr1;  // K=3 column is zero

    // B (4x16): column p = [gx_p, gy_p, 1, 0]. Row-striped: v0 = K0(lanes0-15)/K2(lanes16-31), v1 = K1/K3.
    v2f bm1, bm2;
    bm1.x = hi ? 1.0f : (float)(w0 + lane);        // pixels w0..w0+15
    bm1.y = hi ? 0.0f : (float)h;
    bm2.x = hi ? 1.0f : (float)(w0 + 16 + lane);   // pixels w0+16..w0+31
    bm2.y = bm1.y;

    v8f cz0 = {};
    // D[m][n] = coords component m of pixel n; lanes 0-15 hold N=0..15 with M=vgpr index.
    v8f d1 = __builtin_amdgcn_wmma_f32_16x16x4_f32(false, a, false, bm1, (short)0, cz0, false, false);
    v8f d2 = __builtin_amdgcn_wmma_f32_16x16x4_f32(false, a, false, bm2, (short)0, cz0, false, false);

    // lane p<16 owns pixel w0+p from d1; lane p>=16 owns pixel w0+p from d2's lane p-16
    const float s0 = __shfl(d2[0], lane & 15, 32);
    const float s1 = __shfl(d2[1], lane & 15, 32);
    const float s2 = __shfl(d2[2], lane & 15, 32);
    const float cxn = hi ? s0 : d1[0];
    const float cyn = hi ? s1 : d1[1];
    const float czn = hi ? s2 : d1[2];

    // ---------------- perspective divide, trunc/clamp, bilinear weights ----------------
    const float X = cxn / czn;
    const float Y = cyn / czn;
    const bool valid = (X >= 0.0f) & (X < (float)WP) & (Y >= 0.0f) & (Y < (float)HP);

    // reference: trunc-toward-zero then clamp to [0, WP-2]/[0, HP-2]
    const int x0 = (int)fminf(fmaxf(X, 0.0f), (float)(WP - 2));
    const int y0 = (int)fminf(fmaxf(Y, 0.0f), (float)(HP - 2));
    const float x0f = (float)x0, y0f = (float)y0;
    const float fx0 = X - x0f, fx1 = (x0f + 1.0f) - X;
    const float fy0 = Y - y0f, fy1 = (y0f + 1.0f) - Y;

    // reference's pairing: fx0*fy0 -> (y0,x0), fx0*fy1 -> (y0,x0+1), fx1*fy0 -> (y0+1,x0), fx1*fy1 -> (y0+1,x0+1)
    const float w00 = valid ? fx0 * fy0 : 0.0f;
    const float w01 = valid ? fx0 * fy1 : 0.0f;
    const float w10 = valid ? fx1 * fy0 : 0.0f;
    const float w11 = valid ? fx1 * fy1 : 0.0f;

    // ---------------- 64-channel gather / blend / stream-out ----------------
    const float* __restrict__ p = pv + (size_t)b * CC * PV_PLANE + (size_t)y0 * WP + x0;
    float* __restrict__ o = out + (size_t)b * CC * OUT_PLANE + (size_t)h * WB + w;

#pragma unroll 4
    for (int c = 0; c < CC; ++c) {
        if (c + 2 < CC) {  // cover the 512KB channel-plane stride
            __builtin_prefetch(p + 2 * PV_PLANE, 0, 0);
            __builtin_prefetch(p + 2 * PV_PLANE + WP, 0, 0);
        }
        const float p00 = p[0];
        const float p01 = p[1];
        const float p10 = p[WP];
        const float p11 = p[WP + 1];
        const float v = w00 * p00 + w01 * p01 + w10 * p10 + w11 * p11;
        __builtin_nontemporal_store(v, o);
        p += PV_PLANE;
        o += OUT_PLANE;
    }
}

extern "C" void kernel_launch(void* const* d_in, const int* in_sizes, int n_in,
                              void* d_out, int out_size, void* d_ws, size_t ws_size,
                              hipStream_t stream) {
    (void)in_sizes; (void)n_in; (void)out_size; (void)d_ws; (void)ws_size;
    const float* pv = (const float*)d_in[0];
    const float* rx = (const float*)d_in[1];
    const float* ry = (const float*)d_in[2];
    const float* rz = (const float*)d_in[3];
    float* out = (float*)d_out;

    const int total_pixels = BB * HB * WB;       // 524288
    const int blocks = total_pixels / 256;       // 2048 blocks, block = one (b,h) row
    perspective_warp_kernel<<<blocks, 256, 0, stream>>>(pv, rx, ry, rz, out);
}